// GAT_47132971106798
// MI455X (gfx1250) — compile-verified
//
#include <hip/hip_runtime.h>
#include <hip/hip_bf16.h>
#include <math.h>

typedef __attribute__((ext_vector_type(2))) float v2f;
typedef __attribute__((ext_vector_type(8))) float v8f;

#define ALPHA 0.2f
#define NEG_INF_V -9e15f

// ---------------------------------------------------------------------------
// Tiled f32 WMMA GEMM: C[M,N] = A[M,K] @ B[K,N]
// Block tile 128 x (16*NF), 256 threads = 8 waves. Software-pipelined:
// next K-stage is prefetched into registers while WMMAs consume the current
// LDS stage. B is stored TRANSPOSED in LDS so each lane's (k,k+1) pair is a
// single aligned 64-bit DS load (no repacking movs). Row pad = 20 floats:
// 16B-aligned rows, and 20*dLane mod 64 banks distinct for the 16-lane reads.
// EPI: 0 = none, 1 = ELU, 2 = sigmoid.
// ---------------------------------------------------------------------------
template <int NF, int EPI>
__launch_bounds__(256)
__global__ void wmma_gemm_f32(const float* __restrict__ A,
                              const float* __restrict__ B,
                              float* __restrict__ C,
                              int M, int N, int K)
{
    constexpr int BN  = 16 * NF;
    constexpr int LDA = 20;                 // padded LDS row stride (floats)
    constexpr int NB  = (16 * BN) / 256;    // B elems per thread per stage

    __shared__ __align__(16) float sA[128][LDA];   // sA[m][k]
    __shared__ __align__(16) float sBt[BN][LDA];   // sBt[n][k]  (transposed!)

    const int tid  = threadIdx.x;
    const int lane = tid & 31;
    const int wid  = tid >> 5;
    const int ml   = lane & 15;          // M (A) / N (B,C) index within tile
    const int hi   = lane >> 4;          // lane-group
    const int kk   = hi * 2;             // K offset of this lane group
    const int rowBase = blockIdx.y * 128;
    const int colBase = blockIdx.x * BN;

    v8f acc[NF];
#pragma unroll
    for (int f = 0; f < NF; ++f) acc[f] = (v8f){0.f,0.f,0.f,0.f,0.f,0.f,0.f,0.f};

    // ---- stage load helpers (A: 2 x float4 / thread, B: NB scalars) ----
    auto loadA = [&](int k0, float4* r) {
#pragma unroll
        for (int i = 0; i < 2; ++i) {
            int idx4 = tid + i * 256;                 // 0..511 float4s
            int rr = idx4 >> 2, cc = (idx4 & 3) * 4;  // 128 rows x 16 cols
            r[i] = *(const float4*)&A[(size_t)(rowBase + rr) * K + (k0 + cc)];
        }
    };
    auto loadB = [&](int k0, float* r) {
#pragma unroll
        for (int i = 0; i < NB; ++i) {
            int idx = tid + i * 256;
            int rr = idx / BN, cc = idx % BN;         // 16 k-rows x BN cols
            r[i] = B[(size_t)(k0 + rr) * N + (colBase + cc)];
        }
    };
    auto commit = [&](const float4* ra, const float* rb) {
#pragma unroll
        for (int i = 0; i < 2; ++i) {
            int idx4 = tid + i * 256;
            int rr = idx4 >> 2, cc = (idx4 & 3) * 4;
            *(float4*)&sA[rr][cc] = ra[i];            // 16B-aligned b128 store
        }
#pragma unroll
        for (int i = 0; i < NB; ++i) {
            int idx = tid + i * 256;
            int rr = idx / BN, cc = idx % BN;
            sBt[cc][rr] = rb[i];                      // transpose on the way in
        }
    };

    // ---- prologue: stage 0 ----
    float4 ra[2]; float rb[NB];
    loadA(0, ra); loadB(0, rb);
    commit(ra, rb);
    __syncthreads();

    // ---- pipelined K loop ----
    for (int k0 = 0; k0 < K; k0 += 16) {
        const bool more = (k0 + 16) < K;
        float4 na[2]; float nb[NB];
        if (more) { loadA(k0 + 16, na); loadB(k0 + 16, nb); }  // overlap w/ WMMA

#pragma unroll
        for (int ks = 0; ks < 4; ++ks) {
            const int kb = ks * 4;
            // A 16x4 fragment: contiguous (k,k+1) pair -> one b64 DS load
            v2f a = *(const v2f*)&sA[wid * 16 + ml][kb + kk];
#pragma unroll
            for (int f = 0; f < NF; ++f) {
                v2f b = *(const v2f*)&sBt[f * 16 + ml][kb + kk];
                acc[f] = __builtin_amdgcn_wmma_f32_16x16x4_f32(
                    false, a, false, b, (short)0, acc[f], false, false);
            }
        }
        if (more) {
            __syncthreads();
            commit(na, nb);
            __syncthreads();
        }
    }

    // ---- epilogue: C layout VGPR v -> row v (lanes 0-15) / v+8 (lanes 16-31) ----
#pragma unroll
    for (int f = 0; f < NF; ++f) {
#pragma unroll
        for (int v = 0; v < 8; ++v) {
            int row = rowBase + wid * 16 + v + hi * 8;
            int col = colBase + f * 16 + ml;
            float val = acc[f][v];
            if (EPI == 1)      val = val > 0.f ? val : (__expf(val) - 1.f);  // ELU
            else if (EPI == 2) val = 1.f / (1.f + __expf(-val));             // sigmoid
            C[(size_t)row * N + col] = val;
        }
    }
}

// ---------------------------------------------------------------------------
// Per-row attention logits: e1[i] = Wh[i,:].a[0:F], e2[i] = Wh[i,:].a[F:2F]
// One wave32 per row, shuffle reduction.
// ---------------------------------------------------------------------------
__launch_bounds__(256)
__global__ void evec_kernel(const float* __restrict__ Wh,
                            const float* __restrict__ a,
                            float* __restrict__ e1, float* __restrict__ e2,
                            int F)
{
    const int row  = blockIdx.x * 8 + (threadIdx.x >> 5);
    const int lane = threadIdx.x & 31;
    float s1 = 0.f, s2 = 0.f;
    for (int f = lane; f < F; f += 32) {
        float w = Wh[(size_t)row * F + f];
        s1 += w * a[f];
        s2 += w * a[F + f];
    }
#pragma unroll
    for (int off = 16; off > 0; off >>= 1) {
        s1 += __shfl_down(s1, off, 32);
        s2 += __shfl_down(s2, off, 32);
    }
    if (lane == 0) { e1[row] = s1; e2[row] = s2; }
}

// ---------------------------------------------------------------------------
// Fused masked row softmax (int4/float4 vectorized; adj is the big HBM stream)
// att[i,j] = softmax_j( mask ? lrelu(e1[i]+e2[j]) : -9e15 )
// One 256-thread block per row; the 8192-float row lives in LDS (32 KB).
// ---------------------------------------------------------------------------
__launch_bounds__(256)
__global__ void softmax_row_kernel(const int* __restrict__ adj,
                                   const float* __restrict__ e1,
                                   const float* __restrict__ e2,
                                   float* __restrict__ att,
                                   int N)
{
    __shared__ __align__(16) float srow[8192];
    __shared__ float wred[8];

    const int i    = blockIdx.x;
    const int tid  = threadIdx.x;
    const int lane = tid & 31;
    const int wid  = tid >> 5;
    const float e1i = e1[i];
    const size_t base = (size_t)i * N;

    // pass 1: scores + running max (16B per lane, fully coalesced adj stream)
    float lmax = -3.0e38f;
    for (int j = tid * 4; j < N; j += 1024) {
        int4   av = *(const int4*)&adj[base + j];
        float4 ev = *(const float4*)&e2[j];
        float4 sv;
        {
            float s0 = e1i + ev.x; s0 = s0 > 0.f ? s0 : ALPHA * s0;
            float s1 = e1i + ev.y; s1 = s1 > 0.f ? s1 : ALPHA * s1;
            float s2 = e1i + ev.z; s2 = s2 > 0.f ? s2 : ALPHA * s2;
            float s3 = e1i + ev.w; s3 = s3 > 0.f ? s3 : ALPHA * s3;
            sv.x = (av.x > 0) ? s0 : NEG_INF_V;
            sv.y = (av.y > 0) ? s1 : NEG_INF_V;
            sv.z = (av.z > 0) ? s2 : NEG_INF_V;
            sv.w = (av.w > 0) ? s3 : NEG_INF_V;
        }
        *(float4*)&srow[j] = sv;
        lmax = fmaxf(fmaxf(fmaxf(lmax, sv.x), fmaxf(sv.y, sv.z)), sv.w);
    }
#pragma unroll
    for (int off = 16; off > 0; off >>= 1) lmax = fmaxf(lmax, __shfl_down(lmax, off, 32));
    if (lane == 0) wred[wid] = lmax;
    __syncthreads();
    float rmax = wred[0];
#pragma unroll
    for (int w = 1; w < 8; ++w) rmax = fmaxf(rmax, wred[w]);
    __syncthreads();

    // pass 2: exp + running sum
    float lsum = 0.f;
    for (int j = tid * 4; j < N; j += 1024) {
        float4 sv = *(const float4*)&srow[j];
        sv.x = __expf(sv.x - rmax);
        sv.y = __expf(sv.y - rmax);
        sv.z = __expf(sv.z - rmax);
        sv.w = __expf(sv.w - rmax);
        *(float4*)&srow[j] = sv;
        lsum += (sv.x + sv.y) + (sv.z + sv.w);
    }
#pragma unroll
    for (int off = 16; off > 0; off >>= 1) lsum += __shfl_down(lsum, off, 32);
    if (lane == 0) wred[wid] = lsum;
    __syncthreads();
    float rsum = 0.f;
#pragma unroll
    for (int w = 0; w < 8; ++w) rsum += wred[w];
    const float inv = 1.f / rsum;

    // pass 3: normalize + store (float4 stores)
    for (int j = tid * 4; j < N; j += 1024) {
        float4 sv = *(const float4*)&srow[j];
        sv.x *= inv; sv.y *= inv; sv.z *= inv; sv.w *= inv;
        *(float4*)&att[base + j] = sv;
    }
}

// ---------------------------------------------------------------------------
// Launch sequence
// ---------------------------------------------------------------------------
extern "C" void kernel_launch(void* const* d_in, const int* in_sizes, int n_in,
                              void* d_out, int out_size, void* d_ws, size_t ws_size,
                              hipStream_t stream)
{
    (void)in_sizes; (void)n_in; (void)out_size; (void)ws_size;

    const int N     = 8192;
    const int NFEAT = 512;
    const int NHID  = 256;
    const int NCLS  = 16;

    const float* x  = (const float*)d_in[0];
    const int*   adj= (const int*)  d_in[1];
    const float* W1 = (const float*)d_in[2];
    const float* a1 = (const float*)d_in[3];
    const float* W2 = (const float*)d_in[4];
    const float* a2 = (const float*)d_in[5];

    // output layout: out[8192*16] | h1[8192*256] | att1[8192*8192] | att2[8192*8192]
    float* out  = (float*)d_out;
    float* h1   = out  + (size_t)N * NCLS;
    float* att1 = h1   + (size_t)N * NHID;
    float* att2 = att1 + (size_t)N * N;

    // workspace: Wh1[8192*256] | Wh2[8192*16] | e1a,e2a,e1b,e2b[8192 each]
    float* Wh1 = (float*)d_ws;
    float* Wh2 = Wh1 + (size_t)N * NHID;
    float* e1a = Wh2 + (size_t)N * NCLS;
    float* e2a = e1a + N;
    float* e1b = e2a + N;
    float* e2b = e1b + N;

    // ---- Layer 1 ----
    wmma_gemm_f32<4, 0><<<dim3(NHID / 64, N / 128), 256, 0, stream>>>(x, W1, Wh1, N, NHID, NFEAT);
    evec_kernel<<<N / 8, 256, 0, stream>>>(Wh1, a1, e1a, e2a, NHID);
    softmax_row_kernel<<<N, 256, 0, stream>>>(adj, e1a, e2a, att1, N);
    wmma_gemm_f32<4, 1><<<dim3(NHID / 64, N / 128), 256, 0, stream>>>(att1, Wh1, h1, N, NHID, N);

    // ---- Layer 2 ----
    wmma_gemm_f32<1, 0><<<dim3(NCLS / 16, N / 128), 256, 0, stream>>>(h1, W2, Wh2, N, NCLS, NHID);
    evec_kernel<<<N / 8, 256, 0, stream>>>(Wh2, a2, e1b, e2b, NCLS);
    softmax_row_kernel<<<N, 256, 0, stream>>>(adj, e1b, e2b, att2, N);
    wmma_gemm_f32<1, 2><<<dim3(NCLS / 16, N / 128), 256, 0, stream>>>(att2, Wh2, out, N, NCLS, N);
}